// GatedLinear_10222022165096
// MI455X (gfx1250) — compile-verified
//
#include <hip/hip_runtime.h>
#include <hip/hip_bf16.h>

typedef __attribute__((ext_vector_type(2))) float v2f;
typedef __attribute__((ext_vector_type(8))) float v8f;

#define NTOK 8192
#define DIN  4096
#define DOUT 4096
#define NEXP 8
#define RANK 16
#define RMOE 128

#define BK 32
#define STRD 36   // padded LDS row stride (floats); 36*4B = 144B, 16B aligned, conflict-free

// ---------------------------------------------------------------------------
// CDNA5 async global->LDS copy (16B per lane), tracked by ASYNCcnt.
// ---------------------------------------------------------------------------
__device__ __forceinline__ void async_copy_b128(uint32_t lds_byte_addr, const float* gsrc) {
    asm volatile("global_load_async_to_lds_b128 %0, %1, off"
                 :: "v"(lds_byte_addr), "v"(gsrc)
                 : "memory");
}
__device__ __forceinline__ void wait_async_le12() {
    asm volatile("s_wait_asynccnt 0xc" ::: "memory");
}
__device__ __forceinline__ void wait_async_le0() {
    asm volatile("s_wait_asynccnt 0x0" ::: "memory");
}

// ---------------------------------------------------------------------------
// Issue one stage's copies. Wave grid: WAVES_M x WAVES_N (8 waves total).
// WG tile is (WAVES_M*64) x (WAVES_N*64); each async instr moves 4 rows
// (32 lanes x 16B). Every wave issues exactly 12 copies per stage.
// ---------------------------------------------------------------------------
template <int WAVES_M, int WAVES_N>
__device__ __forceinline__ void issue_stage(uint32_t buf_base,
                                            const float* __restrict__ A, int ldA, int m0,
                                            const float* __restrict__ B, int ldB, int n0,
                                            int kb, int wave, int lane) {
    constexpr int BMt = WAVES_M * 64;
    constexpr int BNt = WAVES_N * 64;
    constexpr int AJ  = BMt / 32;          // async instrs per wave for A tile
    constexpr int BJ  = BNt / 32;          // async instrs per wave for B tile
    const int rsub = lane >> 3;            // 0..3
    const int c4   = (lane & 7) << 2;      // float column of 16B chunk
#pragma unroll
    for (int j = 0; j < AJ; ++j) {
        const int row = wave * (BMt / 8) + j * 4 + rsub;
        async_copy_b128(buf_base + (uint32_t)(row * STRD + c4) * 4u,
                        A + (size_t)(m0 + row) * ldA + kb + c4);
    }
#pragma unroll
    for (int j = 0; j < BJ; ++j) {
        const int row = wave * (BNt / 8) + j * 4 + rsub;
        async_copy_b128(buf_base + (uint32_t)(BMt * STRD * 4) +
                        (uint32_t)(row * STRD + c4) * 4u,
                        B + (size_t)(n0 + row) * ldB + kb + c4);
    }
}

// ---------------------------------------------------------------------------
// One BK-deep stage of fp32 WMMA on the wave's 64x64 patch (4 M x 4 N tiles).
// Tiles stored [row][k], stride STRD.
// ---------------------------------------------------------------------------
__device__ __forceinline__ void mma_stage44(const float* As, const float* Bs,
                                            int wm, int wn, int lm, int hi,
                                            v8f acc[4][4]) {
#pragma unroll
    for (int k4 = 0; k4 < BK; k4 += 4) {
        v2f a[4];
        v2f b[4];
#pragma unroll
        for (int mt = 0; mt < 4; ++mt) {
            // A 16x4: lanes 0-15 hold K={k4,k4+1}, lanes 16-31 K={k4+2,k4+3}
            const float* p = As + (wm * 64 + mt * 16 + lm) * STRD + k4 + 2 * hi;
            a[mt][0] = p[0];
            a[mt][1] = p[1];
        }
#pragma unroll
        for (int nt = 0; nt < 4; ++nt) {
            // B 4x16: VGPR0 = B[k4+hi][n], VGPR1 = B[k4+hi+2][n]; n = lane%16
            const float* p = Bs + (wn * 64 + nt * 16 + lm) * STRD + k4 + hi;
            b[nt][0] = p[0];
            b[nt][1] = p[2];
        }
#pragma unroll
        for (int mt = 0; mt < 4; ++mt) {
#pragma unroll
            for (int nt = 0; nt < 4; ++nt) {
                acc[mt][nt] = __builtin_amdgcn_wmma_f32_16x16x4_f32(
                    false, a[mt], false, b[nt], (short)0, acc[mt][nt], false, false);
            }
        }
    }
}

// ---------------------------------------------------------------------------
// Double-buffered, async-pipelined K loop.
// ---------------------------------------------------------------------------
template <int WAVES_M, int WAVES_N>
__device__ __forceinline__ void gemm_pipeline(float* smem,
                                              const float* __restrict__ A, int ldA, int m0,
                                              const float* __restrict__ B, int ldB, int n0,
                                              int K, int tid, v8f acc[4][4]) {
    constexpr int BMt = WAVES_M * 64;
    constexpr int BNt = WAVES_N * 64;
    constexpr int TILEA = BMt * STRD;                 // floats
    constexpr int BUF   = (BMt + BNt) * STRD;         // floats per pipeline buffer

    const int lane = tid & 31;
    const int wave = tid >> 5;
    const int wm = wave % WAVES_M, wn = wave / WAVES_M;
    const int lm = lane & 15, hi = lane >> 4;
    const uint32_t base = (uint32_t)(uintptr_t)smem;  // LDS byte offset (low 32 bits)
    const int NS = K / BK;

    issue_stage<WAVES_M, WAVES_N>(base, A, ldA, m0, B, ldB, n0, 0, wave, lane);
    if (NS > 1)
        issue_stage<WAVES_M, WAVES_N>(base + BUF * 4, A, ldA, m0, B, ldB, n0, BK, wave, lane);

    for (int i = 0; i < NS; ++i) {
        if (i < NS - 1) wait_async_le12();  // stage i's 12 copies done (in-order)
        else            wait_async_le0();
        __syncthreads();                    // all waves' stage-i data visible
        const float* As = smem + (i & 1) * BUF;
        const float* Bs = As + TILEA;
        mma_stage44(As, Bs, wm, wn, lm, hi, acc);
        __syncthreads();                    // everyone done reading buf i%2
        if (i + 2 < NS)
            issue_stage<WAVES_M, WAVES_N>(base + (uint32_t)((i & 1) * BUF * 4),
                                          A, ldA, m0, B, ldB, n0, (i + 2) * BK, wave, lane);
    }
}

// ---------------------------------------------------------------------------
// Kernel 1: router logits -> top-2 softmax gates (scaled). One wave per token.
// ---------------------------------------------------------------------------
__global__ void __launch_bounds__(256) router_kernel(const float* __restrict__ x,
                                                     const float* __restrict__ rw,
                                                     const float* __restrict__ scalings,
                                                     float* __restrict__ gate_s) {
    const int tok  = blockIdx.x * 8 + (threadIdx.x >> 5);
    const int lane = threadIdx.x & 31;
    const float* xp = x + (size_t)tok * DIN;

    float acc[NEXP];
#pragma unroll
    for (int e = 0; e < NEXP; ++e) acc[e] = 0.0f;

    for (int k = lane; k < DIN; k += 32) {
        const float xv = xp[k];
#pragma unroll
        for (int e = 0; e < NEXP; ++e) acc[e] += xv * rw[e * DIN + k];
    }
#pragma unroll
    for (int e = 0; e < NEXP; ++e) {
#pragma unroll
        for (int off = 16; off > 0; off >>= 1)
            acc[e] += __shfl_xor(acc[e], off, 32);
    }
    int e1 = 0; float v1 = acc[0];
#pragma unroll
    for (int e = 1; e < NEXP; ++e)
        if (acc[e] > v1) { v1 = acc[e]; e1 = e; }
    int e2 = -1; float v2 = -3.4e38f;
#pragma unroll
    for (int e = 0; e < NEXP; ++e)
        if (e != e1 && acc[e] > v2) { v2 = acc[e]; e2 = e; }

    const float g1 = 1.0f / (1.0f + __expf(v2 - v1));
    const float g2 = 1.0f - g1;

    if (lane < NEXP) {
        const float g = (lane == e1) ? g1 : ((lane == e2) ? g2 : 0.0f);
        gate_s[(size_t)tok * NEXP + lane] = g * scalings[lane];
    }
}

// ---------------------------------------------------------------------------
// Kernel 2: h_scaled = (x @ lora_A^T) * gate_s[expert(r)]   [8192 x 128]
// WG tile 256x128 (waves 4x2), grid 32.
// ---------------------------------------------------------------------------
__global__ void __launch_bounds__(256) lora_h_kernel(const float* __restrict__ x,
                                                     const float* __restrict__ lora_A,
                                                     const float* __restrict__ gate_s,
                                                     float* __restrict__ h_scaled) {
    __shared__ __align__(16) float smem[2 * (256 + 128) * STRD];

    const int tid  = threadIdx.x;
    const int m0   = blockIdx.x * 256;
    const int lane = tid & 31;
    const int wave = tid >> 5;
    const int wm = wave % 4, wn = wave / 4;
    const int lm = lane & 15, hi = lane >> 4;

    v8f acc[4][4];
#pragma unroll
    for (int mt = 0; mt < 4; ++mt)
#pragma unroll
        for (int nt = 0; nt < 4; ++nt)
            acc[mt][nt] = (v8f){};

    gemm_pipeline<4, 2>(smem, x, DIN, m0, lora_A, DIN, 0, DIN, tid, acc);

#pragma unroll
    for (int mt = 0; mt < 4; ++mt) {
        const int mbase = m0 + wm * 64 + mt * 16 + 8 * hi;
#pragma unroll
        for (int nt = 0; nt < 4; ++nt) {
            const int r = wn * 64 + nt * 16 + lm;   // rank index 0..127
            const int ex = r >> 4;                  // expert = r / RANK
#pragma unroll
            for (int v = 0; v < 8; ++v) {
                const int m = mbase + v;
                const float sc = gate_s[(size_t)m * NEXP + ex];
                h_scaled[(size_t)m * RMOE + r] = acc[mt][nt][v] * sc;
            }
        }
    }
}

// ---------------------------------------------------------------------------
// Kernel 3: out = x @ base_w^T + base_b + h_scaled @ lora_B^T  (fused K loop)
// WG tile 128x256 (waves 2x4), grid (16, 64).
// ---------------------------------------------------------------------------
__global__ void __launch_bounds__(256) main_gemm_kernel(const float* __restrict__ x,
                                                        const float* __restrict__ base_w,
                                                        const float* __restrict__ base_b,
                                                        const float* __restrict__ lora_B,
                                                        const float* __restrict__ h_scaled,
                                                        float* __restrict__ out) {
    __shared__ __align__(16) float smem[2 * (128 + 256) * STRD];

    const int tid  = threadIdx.x;
    const int m0   = blockIdx.y * 128;
    const int n0   = blockIdx.x * 256;
    const int lane = tid & 31;
    const int wave = tid >> 5;
    const int wm = wave % 2, wn = wave / 2;
    const int lm = lane & 15, hi = lane >> 4;

    v8f acc[4][4];
#pragma unroll
    for (int mt = 0; mt < 4; ++mt)
#pragma unroll
        for (int nt = 0; nt < 4; ++nt)
            acc[mt][nt] = (v8f){};

    // Phase 1: base GEMM, K = 4096
    gemm_pipeline<2, 4>(smem, x, DIN, m0, base_w, DIN, n0, DIN, tid, acc);
    // Phase 2: LoRA-B, K = 128 (gate/scaling already folded into h_scaled)
    gemm_pipeline<2, 4>(smem, h_scaled, RMOE, m0, lora_B, RMOE, n0, RMOE, tid, acc);

#pragma unroll
    for (int mt = 0; mt < 4; ++mt) {
        const int mbase = m0 + wm * 64 + mt * 16 + 8 * hi;
#pragma unroll
        for (int nt = 0; nt < 4; ++nt) {
            const int n = n0 + wn * 64 + nt * 16 + lm;
            const float bias = base_b[n];
#pragma unroll
            for (int v = 0; v < 8; ++v) {
                const int m = mbase + v;
                out[(size_t)m * DOUT + n] = acc[mt][nt][v] + bias;
            }
        }
    }
}

// ---------------------------------------------------------------------------
extern "C" void kernel_launch(void* const* d_in, const int* in_sizes, int n_in,
                              void* d_out, int out_size, void* d_ws, size_t ws_size,
                              hipStream_t stream) {
    const float* x        = (const float*)d_in[0];
    const float* base_w   = (const float*)d_in[1];
    const float* base_b   = (const float*)d_in[2];
    const float* lora_A   = (const float*)d_in[3];
    const float* lora_B   = (const float*)d_in[4];
    const float* router_w = (const float*)d_in[5];
    const float* scalings = (const float*)d_in[6];
    // d_in[7] = top_k (=2); top-2 is specialized in router_kernel.

    float* gate_s   = (float*)d_ws;                 // [NTOK, NEXP]
    float* h_scaled = gate_s + (size_t)NTOK * NEXP; // [NTOK, RMOE]
    float* out      = (float*)d_out;

    router_kernel<<<NTOK / 8, 256, 0, stream>>>(x, router_w, scalings, gate_s);
    lora_h_kernel<<<NTOK / 256, 256, 0, stream>>>(x, lora_A, gate_s, h_scaled);

    dim3 grid(DOUT / 256, NTOK / 128);
    main_gemm_kernel<<<grid, 256, 0, stream>>>(x, base_w, base_b, lora_B, h_scaled, out);
}